// LengthRegulator_764504179410
// MI455X (gfx1250) — compile-verified
//
#include <hip/hip_runtime.h>
#include <stdint.h>

// LengthRegulator (alpha=1.0) for MI455X / gfx1250.
// B=256 rows, T=2048 durations/row, max_len=24576 output frames/row.
// Store-bandwidth bound (~25 MB stores => ~1.2us at 23.3 TB/s). Strategy:
//   - async global->LDS staging of the duration row (ASYNCcnt path)
//   - wave32 shuffle scan for the masked cumulative sum
//   - branchless 11-step binary search in LDS per output frame
//   - coalesced int4 (b128) stores of the mel2ph map

#define T_LEN   2048
#define TPB     256
#define PER_THR (T_LEN / TPB)   // 8
#define SPLITS  4

__global__ __launch_bounds__(TPB)
void lenreg_kernel(const int* __restrict__ ds,
                   const int* __restrict__ ilens,
                   int* __restrict__ out,
                   int max_len)
{
    __shared__ int s_cs[T_LEN];        // raw durations, then inclusive cumsum
    __shared__ int s_woff[TPB / 32];   // per-wave exclusive offsets
    __shared__ int s_total;

    const int b     = blockIdx.x;
    const int slice = blockIdx.y;
    const int tid   = threadIdx.x;
    const int lane  = tid & 31;
    const int wave  = tid >> 5;
    const int ilen  = ilens[b];

    // ---------------- Phase 1: async copy of the duration row into LDS ------
    // gfx1250 GLOBAL_LOAD_ASYNC_TO_LDS_B32 (GV mode), tracked by ASYNCcnt.
    const int* grow = ds + (size_t)b * T_LEN;
#pragma unroll
    for (int k = 0; k < PER_THR; ++k) {
        const int idx       = k * TPB + tid;
        unsigned  lds_off   = (unsigned)(uintptr_t)(&s_cs[idx]);
        uint64_t  gaddr     = (uint64_t)(uintptr_t)(grow + idx);
        asm volatile("global_load_async_to_lds_b32 %0, %1, off"
                     :: "v"(lds_off), "v"(gaddr)
                     : "memory");
    }
    asm volatile("s_wait_asynccnt 0" ::: "memory");
    __syncthreads();

    // ---------------- Phase 2: masked inclusive scan -------------------------
    // Each thread owns the contiguous chunk [tid*8, tid*8+8).
    const int base = tid * PER_THR;
    int v[PER_THR];
    int run = 0;
#pragma unroll
    for (int j = 0; j < PER_THR; ++j) {
        const int idx = base + j;
        const int d   = (idx < ilen) ? s_cs[idx] : 0;
        run += d;
        v[j] = run;            // local inclusive prefix
    }

    // wave32 inclusive scan of per-thread sums (Hillis-Steele via shuffles)
    int x = run;
#pragma unroll
    for (int sh = 1; sh < 32; sh <<= 1) {
        int y = __shfl_up(x, sh, 32);
        if (lane >= sh) x += y;
    }
    if (lane == 31) s_woff[wave] = x;  // wave total
    __syncthreads();
    if (tid == 0) {
        int acc = 0;
        for (int w = 0; w < TPB / 32; ++w) {
            int t = s_woff[w];
            s_woff[w] = acc;           // exclusive wave offset
            acc += t;
        }
        s_total = acc;
    }
    __syncthreads();

    const int excl = s_woff[wave] + (x - run);   // thread-exclusive offset
#pragma unroll
    for (int j = 0; j < PER_THR; ++j)
        s_cs[base + j] = excl + v[j];            // inclusive cumsum in LDS
    __syncthreads();

    int total = s_total;
    if (total == 0) {
        // Faithful fallback: all-zero durations -> d=1 on valid positions,
        // i.e. cs[i] = min(i+1, ilen) analytically.
#pragma unroll
        for (int j = 0; j < PER_THR; ++j) {
            const int idx = base + j;
            s_cs[idx] = (idx + 1 < ilen) ? (idx + 1) : ilen;
        }
        total = ilen;
        __syncthreads();
    }

    // ---------------- Phase 3: searchsorted + coalesced b128 stores ----------
    const int slice_len = max_len / SPLITS;            // 6144
    const int iters     = slice_len / (TPB * 4);       // 6
    int* orow = out + (size_t)b * max_len + (size_t)slice * slice_len;

    for (int k = 0; k < iters; ++k) {
        const int m0 = (k * TPB + tid) * 4;            // offset within slice
        const int mm = slice * slice_len + m0;         // absolute frame index
        int rr[4];
#pragma unroll
        for (int j = 0; j < 4; ++j) {
            const int m = mm + j;
            int val = 0;
            if (m < total) {
                // branchless upper_bound over 2048-entry sorted cs in LDS
                int lo = 0;
#pragma unroll
                for (int step = T_LEN >> 1; step >= 1; step >>= 1)
                    lo += (s_cs[lo + step - 1] <= m) ? step : 0;
                val = lo + 1;
            }
            rr[j] = val;
        }
        if (mm + 3 < max_len) {
            int4 r; r.x = rr[0]; r.y = rr[1]; r.z = rr[2]; r.w = rr[3];
            *(int4*)(orow + m0) = r;                   // global_store_b128
        } else {
            for (int j = 0; j < 4; ++j)
                if (mm + j < max_len) orow[m0 + j] = rr[j];
        }
    }
}

extern "C" void kernel_launch(void* const* d_in, const int* in_sizes, int n_in,
                              void* d_out, int out_size, void* d_ws, size_t ws_size,
                              hipStream_t stream) {
    (void)d_ws; (void)ws_size; (void)n_in;
    const int* ds    = (const int*)d_in[0];   // (B, T) int32
    const int* ilens = (const int*)d_in[1];   // (B,)   int32
    int*       out   = (int*)d_out;           // (B, max_len) int32

    const int B       = in_sizes[1];          // 256
    const int max_len = out_size / B;         // 24576

    dim3 grid(B, SPLITS);
    dim3 block(TPB);
    lenreg_kernel<<<grid, block, 0, stream>>>(ds, ilens, out, max_len);
}